// HandBallPrimitiveRender_50689204027923
// MI455X (gfx1250) — compile-verified
//
#include <hip/hip_runtime.h>
#include <math.h>

// ---------------------------------------------------------------------------
// HandBallPrimitiveRender for MI455X (gfx1250, wave32)
//
// Roofline: output = 236MB part_maps + 4.9MB depth_maps, written once, never
// re-read. At 23.3 TB/s HBM the floor is ~10.4us. Arithmetic is ~25 VALU +
// 4 v_sqrt per 16B stored -> pure store-bandwidth bound. Therefore:
//   * f32 everywhere (precision is free, bytes are what matter)
//   * b128 non-temporal stores (236MB > 192MB L2, streaming)
//   * coalesced: lane i stores 16B at consecutive addresses
// Skinning (einsum bvij,vj->bvi) is K=4 dot products -> V_WMMA_F32_16X16X4_F32
// (4 vertices x 4 rows packed into the 16x4 A operand per wave).
// ---------------------------------------------------------------------------

typedef float v2f __attribute__((ext_vector_type(2)));
typedef float v4f __attribute__((ext_vector_type(4)));
typedef float v8f __attribute__((ext_vector_type(8)));

#define W_PIX   320
#define H_PIX   240
#define N_BONES 16
#define N_VERTS 48
#define N_BATCH 16
#define BG_DEPTH  100.0f
#define CLAMP_MIN 0.01f

// ---------------------------------------------------------------------------
// Kernel 1: vertex skinning via V_WMMA_F32_16X16X4_F32.
// One wave per batch b. Each WMMA handles 4 vertices:
//   A (16x4): row m = T[b, bone(vg*4 + m>>2)][m&3][:]   (m = 4*dv + comp)
//   B (4x16): col n = vertices[vg*4 + (n>>2)][:]        (cols repeat per dv)
//   D[m,n] = T_row_m . vert_{n>>2}  -> valid where (m>>2)==(n>>2)
// A-operand VGPR layout (ISA 7.12.2, 32-bit A 16x4): lanes 0-15 carry K=0,1
// in v0,v1; lanes 16-31 carry K=2,3. B mirrors this with N across lanes.
// D layout: VGPR r, lanes 0-15 -> D[r, lane]; lanes 16-31 -> D[r+8, lane-16].
//
// Extraction: the four writer lanes hold a whole output vertex contiguously:
//   lane 0  : D[0..3,   0] = d[0..3] -> vertex vg*4+0
//   lane 4  : D[4..7,   4] = d[4..7] -> vertex vg*4+1
//   lane 24 : D[8..11,  8] = d[0..3] -> vertex vg*4+2
//   lane 28 : D[12..15,12] = d[4..7] -> vertex vg*4+3
// => one predicated global_store_b128 per WMMA.
// ---------------------------------------------------------------------------
__global__ void __launch_bounds__(32)
skin_wmma_kernel(const float* __restrict__ T,        // [B,16,4,4]
                 const float* __restrict__ verts,    // [48,4] (w==1)
                 const int*   __restrict__ bone_idx, // [48]
                 float*       __restrict__ skinned)  // [B,48,4] (workspace)
{
    const int b    = blockIdx.x;
    const int lane = threadIdx.x;        // 0..31, EXEC all-ones at the WMMA
    const int m    = lane & 15;          // A row / B col index
    const int kb   = (lane >> 4) << 1;   // K base: 0 (lanes 0-15) or 2 (16-31)
    const int dv   = m >> 2;             // vertex-in-group 0..3
    const int comp = m & 3;              // matrix row (output component)

    const bool hiHalf = (lane & 4) != 0;                         // use d[4..7]
    const int  gOut   = ((lane >> 4) ? 2 : 0) + (hiHalf ? 1 : 0); // out vertex-in-group
    const bool writer = (lane == 0) || (lane == 4) || (lane == 24) || (lane == 28);

    for (int vg = 0; vg < N_VERTS / 4; ++vg) {
        const int vtx  = vg * 4 + dv;
        const int bone = bone_idx[vtx];

        const float* trow = T + (((size_t)(b * N_BONES + bone) * 4) + comp) * 4;
        v2f a;  a.x = trow[kb];              a.y = trow[kb + 1];
        v2f bm; bm.x = verts[vtx * 4 + kb];  bm.y = verts[vtx * 4 + kb + 1];

        v8f c = {};
        v8f d = __builtin_amdgcn_wmma_f32_16x16x4_f32(
            /*neg_a=*/false, a, /*neg_b=*/false, bm,
            /*c_mod=*/(short)0, c, /*reuse_a=*/false, /*reuse_b=*/false);

        v4f out;
        out.x = hiHalf ? d[4] : d[0];
        out.y = hiHalf ? d[5] : d[1];
        out.z = hiHalf ? d[6] : d[2];
        out.w = hiHalf ? d[7] : d[3];

        if (writer)
            *(v4f*)(skinned + ((size_t)b * N_VERTS + vg * 4 + gOut) * 4) = out;
    }
}

// ---------------------------------------------------------------------------
// Kernel 2: streaming splat render. Store-bandwidth bound.
// Grid (H_PIX/4, N_BATCH), 320 threads = 10 waves. Each block: 4 rows of one
// batch; each thread: 4 consecutive pixels -> one NT b128 store per vertex.
// ---------------------------------------------------------------------------
__global__ void __launch_bounds__(320)
render_kernel(const float* __restrict__ skinned,  // [B,48,4]
              const float* __restrict__ radiuses, // [48]
              float*       __restrict__ part,     // [B,48,H,W]
              float*       __restrict__ depth)    // [B,H,W]
{
    __shared__ v4f sv[N_VERTS];   // x, y, z, r^2

    const int tid = threadIdx.x;  // 0..319
    const int b   = blockIdx.y;

    if (tid < N_VERTS) {
        const float* s = skinned + ((size_t)b * N_VERTS + tid) * 4;
        const float  r = radiuses[tid];
        v4f t; t.x = s[0]; t.y = s[1]; t.z = s[2]; t.w = r * r;
        sv[tid] = t;
    }
    __syncthreads();

    const int row = tid / 80;          // 0..3
    const int q   = tid - row * 80;    // 0..79
    const int w0  = q * 4;
    const int h   = blockIdx.x * 4 + row;

    const float yg = ((float)h - (float)(H_PIX / 2)) * (300.0f / (float)H_PIX);
    float xg[4];
    #pragma unroll
    for (int i = 0; i < 4; ++i)
        xg[i] = ((float)(w0 + i) - (float)(W_PIX / 2)) * (300.0f / (float)W_PIX);

    v4f mn;
    mn.x = mn.y = mn.z = mn.w = __builtin_huge_valf();

    const size_t row_off = (size_t)h * W_PIX + w0;
    float* pbase = part + (size_t)b * N_VERTS * H_PIX * W_PIX + row_off;

    for (int v = 0; v < N_VERTS; ++v) {
        const v4f s  = sv[v];                       // broadcast ds_load_b128
        const float dy = yg - s.y;
        const float c1 = s.w - dy * dy;             // r^2 - dy^2 (per vertex)

        v4f val;
        #pragma unroll
        for (int i = 0; i < 4; ++i) {
            const float dx = xg[i] - s.x;
            const float t  = c1 - dx * dx;          // r^2 - dx^2 - dy^2
            const float sq = fmaxf(t, CLAMP_MIN);   // clip(..., 0.01)
            val[i] = (sq != CLAMP_MIN) ? (s.z - __builtin_amdgcn_sqrtf(sq))
                                       : BG_DEPTH;
        }

        __builtin_nontemporal_store(val, (v4f*)(pbase + (size_t)v * H_PIX * W_PIX));

        mn.x = fminf(mn.x, val.x);
        mn.y = fminf(mn.y, val.y);
        mn.z = fminf(mn.z, val.z);
        mn.w = fminf(mn.w, val.w);
    }

    __builtin_nontemporal_store(mn, (v4f*)(depth + (size_t)b * H_PIX * W_PIX + row_off));
}

// ---------------------------------------------------------------------------
extern "C" void kernel_launch(void* const* d_in, const int* in_sizes, int n_in,
                              void* d_out, int out_size, void* d_ws, size_t ws_size,
                              hipStream_t stream) {
    const float* T     = (const float*)d_in[0];  // [16,16,4,4]
    const float* verts = (const float*)d_in[1];  // [48,4]
    const float* rad   = (const float*)d_in[2];  // [48]
    const int*   bidx  = (const int*)d_in[3];    // [48]

    float* part  = (float*)d_out;                                    // [16,48,240,320]
    float* depth = part + (size_t)N_BATCH * N_VERTS * H_PIX * W_PIX; // [16,240,320]
    float* skinned = (float*)d_ws;                                   // [16,48,4] = 12 KB

    skin_wmma_kernel<<<dim3(N_BATCH), dim3(32), 0, stream>>>(T, verts, bidx, skinned);
    render_kernel<<<dim3(H_PIX / 4, N_BATCH), dim3(320), 0, stream>>>(skinned, rad, part, depth);
}